// SoftFaceColorShader_10342281249253
// MI455X (gfx1250) — compile-verified
//
#include <hip/hip_runtime.h>
#include <hip/hip_bf16.h>

// SoftFaceColorShader: pure streaming/gather blend. Memory-bound (~300 MB/call
// -> ~13us floor at 23.3 TB/s). No matrix structure -> no WMMA. CDNA5 paths
// used: NT temporal hints (th:TH_LOAD_NT) on once-read streams so L2 keeps the
// reused 1.9 MB face_colors table hot; staged wide loads (deep s_clause
// batches, one wait point); fully branchless math (no exec divergence).

#define KLAYERS 10
#define DCH     3

// Native clang vectors (HIP's vector classes are rejected by the nontemporal
// builtins).
typedef __attribute__((ext_vector_type(2))) int   v2i;
typedef __attribute__((ext_vector_type(2))) float v2f;
typedef __attribute__((ext_vector_type(4))) float v4f;

__global__ __launch_bounds__(256) void soft_face_color_shader(
    const int*   __restrict__ pix_to_face,   // [P, K]   (40 B/px, 8B aligned)
    const float* __restrict__ bary,          // [P, K, 3](120 B/px, 8B aligned)
    const float* __restrict__ zbuf,          // [P, K]
    const float* __restrict__ dists,         // [P, K]
    const float* __restrict__ face_colors,   // [N*F, 3] (1.9 MB, L2-resident)
    v4f*         __restrict__ out,           // [P] (D+1 = 4 channels)
    int P)
{
    const int tid = blockIdx.x * blockDim.x + threadIdx.x;
    if (tid >= P) return;

    const long base  = (long)tid * KLAYERS;
    const long baseb = base * 3;

    const float inv_sigma = 1.0f / 1e-4f;   // 1/SIGMA
    const float inv_gamma = 1.0f / 1e-2f;   // 1/GAMMA
    const float inv_zr    = 1.0f / 99.0f;   // 1/(ZFAR-ZNEAR)
    const float eps       = 1e-10f;

    // ---- stage 1: all pix_to_face indices up front (b128/b128/b64, NT) ----
    int fidx[KLAYERS];
    {
        const v2i* p = (const v2i*)(pix_to_face + base);
#pragma unroll
        for (int j = 0; j < KLAYERS / 2; ++j) {
            const v2i t = __builtin_nontemporal_load(p + j);
            fidx[2 * j]     = t.x;
            fidx[2 * j + 1] = t.y;
        }
    }

    // ---- stage 2: issue all 10 gathers, branchless + mergeable ----
    // v_max_i32 clamp, unsigned zero-extended base offset (no 64-bit select,
    // no exec branches); common pointer fp so fp[0..2] merges into one b96.
    float fc0[KLAYERS], fc1[KLAYERS], fc2[KLAYERS];
#pragma unroll
    for (int k = 0; k < KLAYERS; ++k) {
        const int idxm = (fidx[k] < 0) ? 0 : fidx[k];     // v_max_i32
        const float* fp = face_colors + (unsigned)(idxm * DCH);
        fc0[k] = fp[0];
        fc1[k] = fp[1];
        fc2[k] = fp[2];
    }

    // ---- stage 3: stream zbuf / dists / bary (wide, NT) ----
    float zb[KLAYERS], dd[KLAYERS], bl[KLAYERS * 3];
    {
        const v2f* pz = (const v2f*)(zbuf + base);
        const v2f* pd = (const v2f*)(dists + base);
        const v2f* pb = (const v2f*)(bary + baseb);
#pragma unroll
        for (int j = 0; j < KLAYERS / 2; ++j) {
            const v2f tz = __builtin_nontemporal_load(pz + j);
            zb[2 * j] = tz.x; zb[2 * j + 1] = tz.y;
            const v2f td = __builtin_nontemporal_load(pd + j);
            dd[2 * j] = td.x; dd[2 * j + 1] = td.y;
        }
#pragma unroll
        for (int j = 0; j < (KLAYERS * 3) / 2; ++j) {
            const v2f tb = __builtin_nontemporal_load(pb + j);
            bl[2 * j] = tb.x; bl[2 * j + 1] = tb.y;
        }
    }

    // ---- stage 4: per-layer math, fully branchless (mask as 0/1 float) ----
    float prob[KLAYERS], zinv[KLAYERS];
    float c0[KLAYERS], c1[KLAYERS], c2[KLAYERS];
    float one_minus_prod = 1.0f;
    float zmax = 0.0f;
#pragma unroll
    for (int k = 0; k < KLAYERS; ++k) {
        const float mk = (fidx[k] < 0) ? 0.0f : 1.0f;   // v_cndmask

        // vert colors are the face color repeated for 3 verts:
        // colors[d] = (b0+b1+b2) * face_color[d]
        const float s = (bl[3 * k] + bl[3 * k + 1] + bl[3 * k + 2]) * mk;

        // prob = sigmoid(-d/sigma) * mask
        const float p  = mk / (1.0f + __expf(dd[k] * inv_sigma));
        const float zi = mk * ((100.0f - zb[k]) * inv_zr);

        prob[k] = p;
        zinv[k] = zi;
        c0[k]   = s * fc0[k];
        c1[k]   = s * fc1[k];
        c2[k]   = s * fc2[k];

        one_minus_prod *= (1.0f - p);
        zmax = fmaxf(zmax, zi);
    }
    zmax = fmaxf(zmax, eps);   // z_inv_max = max(max_k z_inv, EPS)

    // ---- stage 5: inverse-depth softmax blend (all in registers) ----
    float sumw = 0.0f, a0 = 0.0f, a1 = 0.0f, a2 = 0.0f;
#pragma unroll
    for (int k = 0; k < KLAYERS; ++k) {
        const float w = prob[k] * __expf((zinv[k] - zmax) * inv_gamma);
        sumw += w;
        a0 += w * c0[k];
        a1 += w * c1[k];
        a2 += w * c2[k];
    }

    const float delta = fmaxf(__expf((eps - zmax) * inv_gamma), eps);
    const float inv_denom = 1.0f / (sumw + delta);

    v4f o;
    o.x = a0 * inv_denom;
    o.y = a1 * inv_denom;
    o.z = a2 * inv_denom;
    o.w = 1.0f - one_minus_prod;   // alpha
    __builtin_nontemporal_store(o, out + tid);
}

extern "C" void kernel_launch(void* const* d_in, const int* in_sizes, int n_in,
                              void* d_out, int out_size, void* d_ws, size_t ws_size,
                              hipStream_t stream) {
    (void)n_in; (void)d_ws; (void)ws_size; (void)out_size;

    const int*   pix_to_face = (const int*)  d_in[0];  // [N,H,W,K]
    const float* bary        = (const float*)d_in[1];  // [N,H,W,K,3]
    const float* zbuf        = (const float*)d_in[2];  // [N,H,W,K]
    const float* dists       = (const float*)d_in[3];  // [N,H,W,K]
    const float* face_colors = (const float*)d_in[4];  // [N,F,D] flat [N*F,3]
    v4f*         out         = (v4f*)d_out;            // [N,H,W,4]

    const int P = in_sizes[0] / KLAYERS;               // N*H*W pixels

    const int block = 256;                             // 8 wave32 waves
    const int grid  = (P + block - 1) / block;
    soft_face_color_shader<<<grid, block, 0, stream>>>(
        pix_to_face, bary, zbuf, dists, face_colors, out, P);
}